// color_invariant_triplet_9345848836728
// MI455X (gfx1250) — compile-verified
//
#include <hip/hip_runtime.h>

// CDNA5 / gfx1250, wave32.
typedef __attribute__((ext_vector_type(2))) float v2f;
typedef __attribute__((ext_vector_type(8))) float v8f;

#define IN_FEATS 64

// One wave (32 lanes) handles 16 h-edges x 64 feats.
// out[e] = table[combo(e)] realized as onehot8 x table via two chained
// V_WMMA_F32_16X16X4_F32 (K=4 covering combos 0-3, then 4-7). All products
// are exact (0/1 * f32), so the result is bit-identical to the reference's
// (e1+e2)+e3 per-combo sum.
__global__ __launch_bounds__(256) void color_invariant_triplet_kernel(
    const int* __restrict__ h_nodes,
    const int* __restrict__ src_g,
    const int* __restrict__ dst_g,
    const int* __restrict__ src_h,
    const int* __restrict__ dst_h,
    const float* __restrict__ e1_w,
    const float* __restrict__ e2_w,
    const float* __restrict__ e3_w,
    float* __restrict__ out,
    int n_eh)
{
    // 8 combos x 64 feats = 2 KB combo table in LDS, same add order as reference.
    __shared__ float table[8][IN_FEATS];
    for (int i = threadIdx.x; i < 8 * IN_FEATS; i += 256) {
        const int c = i >> 6;
        const int f = i & 63;
        float v = e1_w[(c & 1) * IN_FEATS + f] + e2_w[((c >> 1) & 1) * IN_FEATS + f];
        v = v + e3_w[((c >> 2) & 1) * IN_FEATS + f];
        table[c][f] = v;
    }
    __syncthreads();

    const int lane = threadIdx.x & 31;
    const int wave = threadIdx.x >> 5;
    const int lo   = lane & 15;   // N index / edge-in-tile
    const int hiS  = lane >> 4;   // 0: lanes 0-15, 1: lanes 16-31
    const int r0   = hiS * 2;     // K base for this half-wave (A layout: K0,K1 | K2,K3)

    // B operands for the 4 feature tiles (B 4x16 f32: VGPR0 = rows {0,2},
    // VGPR1 = rows {1,3}, lane halves select row vs row+2).
    v2f b1[4], b2[4];
#pragma unroll
    for (int t = 0; t < 4; ++t) {
        const int col = t * 16 + lo;
        b1[t].x = table[r0 + 0][col];
        b1[t].y = table[r0 + 1][col];
        b2[t].x = table[r0 + 4][col];
        b2[t].y = table[r0 + 5][col];
    }

    const int base = blockIdx.x * 128 + wave * 16;   // 16 edges per wave
    const int e    = base + lo;
    const int ec   = (e < n_eh) ? e : (n_eh - 1);    // clamp: keep EXEC all-1s for WMMA

    // Triple-gather for this lane's edge (lanes l and l+16 read the same
    // addresses -> coalesced/broadcast; all gathered arrays are L2-resident).
    const int sh = src_h[ec];
    const int dh = dst_h[ec];
    const int ha = h_nodes[src_g[sh]];
    const int hb = h_nodes[dst_g[sh]];
    const int hc = h_nodes[dst_g[dh]];
    const int combo = (ha == hc ? 1 : 0) | (ha == hb ? 2 : 0) | (hc == hb ? 4 : 0);

    // A operands: one-hot rows. Lane half selects which K pair it holds.
    v2f a1, a2;
    a1.x = (combo == r0 + 0) ? 1.0f : 0.0f;
    a1.y = (combo == r0 + 1) ? 1.0f : 0.0f;
    a2.x = (combo == r0 + 4) ? 1.0f : 0.0f;
    a2.y = (combo == r0 + 5) ? 1.0f : 0.0f;

    // All 4 feature tiles: D = A1*B1 + (A2*B2 + 0) == table[combo] exactly.
    v8f acc[4];
#pragma unroll
    for (int t = 0; t < 4; ++t) {
        v8f c0 = {};
        c0 = __builtin_amdgcn_wmma_f32_16x16x4_f32(
            false, a2, false, b2[t], (short)0, c0, false, false);
        acc[t] = __builtin_amdgcn_wmma_f32_16x16x4_f32(
            false, a1, false, b1[t], (short)0, c0, false, false);
    }

    // D layout: VGPR j holds (M = j + 8*hiS, N = lo).
    if (base + 16 <= n_eh) {
        // Wave-uniform fast path: one per-lane base pointer, 32 stores with
        // immediate offsets (j*256 + t*64 bytes), nontemporal (output is
        // write-once; keep L2 for the gather arrays).
        float* p = out + (size_t)(base + hiS * 8) * IN_FEATS + lo;
#pragma unroll
        for (int t = 0; t < 4; ++t)
#pragma unroll
            for (int j = 0; j < 8; ++j)
                __builtin_nontemporal_store(acc[t][j], p + j * IN_FEATS + t * 16);
    } else {
        // Partial or past-end waves (never taken when n_eh % 128 == 0).
#pragma unroll
        for (int t = 0; t < 4; ++t)
#pragma unroll
            for (int j = 0; j < 8; ++j) {
                const int row = base + j + hiS * 8;
                if (row < n_eh)
                    out[(size_t)row * IN_FEATS + t * 16 + lo] = acc[t][j];
            }
    }
}

extern "C" void kernel_launch(void* const* d_in, const int* in_sizes, int n_in,
                              void* d_out, int out_size, void* d_ws, size_t ws_size,
                              hipStream_t stream) {
    const int*   h_nodes = (const int*)d_in[0];
    const int*   src_g   = (const int*)d_in[1];
    const int*   dst_g   = (const int*)d_in[2];
    const int*   src_h   = (const int*)d_in[3];
    const int*   dst_h   = (const int*)d_in[4];
    const float* e1_w    = (const float*)d_in[5];
    const float* e2_w    = (const float*)d_in[6];
    const float* e3_w    = (const float*)d_in[7];
    float* out = (float*)d_out;

    const int n_eh = in_sizes[3];                 // E_H (src_h element count)
    const int blocks = (n_eh + 127) / 128;        // 128 edges per 256-thread block
    hipLaunchKernelGGL(color_invariant_triplet_kernel,
                       dim3(blocks), dim3(256), 0, stream,
                       h_nodes, src_g, dst_g, src_h, dst_h,
                       e1_w, e2_w, e3_w, out, n_eh);
}